// StridedFourier_56925496541936
// MI455X (gfx1250) — compile-verified
//
#include <hip/hip_runtime.h>

typedef float v2f __attribute__((ext_vector_type(2)));
typedef float v8f __attribute__((ext_vector_type(8)));

constexpr int KSIZE  = 1024;
constexpr int STRIDE = 256;
constexpr int BATCH  = 32;
constexpr int TIME   = 262144;
constexpr int N_WIN  = (TIME - KSIZE) / STRIDE + 1;          // 1021
constexpr int WPB    = 4;                                     // windows (waves) per block
constexpr int SEG    = WPB * STRIDE + (KSIZE - STRIDE);       // 1792 floats
constexpr int GROUPS = (N_WIN + WPB - 1) / WPB;               // 256

#define WMMA_F32(a, b, c) \
    __builtin_amdgcn_wmma_f32_16x16x4_f32(false, (a), false, (b), (short)0, (c), false, false)

__global__ __launch_bounds__(128)
void StridedFourier_kernel(const float* __restrict__ x, float* __restrict__ out)
{
    __shared__ float Cm[1024];          // cos(2*pi*r*c/32), 32x32 row-major (symmetric)
    __shared__ float Sm[1024];          // sin(2*pi*r*c/32)                 (symmetric)
    __shared__ v2f   Tw[1024];          // twiddle table {cos, sin}(2*pi*p/1024)
    __shared__ float seg[SEG];          // shared input segment (4 overlapping windows)
    __shared__ float Ar[WPB][1024];     // per-wave stage-1 real part [k1][n2]
    __shared__ float Ai[WPB][1024];     // per-wave stage-1 imag part [k1][n2]

    const int tid = threadIdx.x;
    const int g   = blockIdx.x % GROUPS;     // window group
    const int b   = blockIdx.x / GROUPS;     // batch row

    // ---- DFT-32 basis matrices + twiddle table (once per block) ----
    for (int i = tid; i < 1024; i += 128) {
        const int r = i >> 5, c = i & 31;
        const float ang = (float)(r * c) * (6.283185307179586f / 32.0f);
        Cm[i] = cosf(ang);
        Sm[i] = sinf(ang);
        float s, cc;
        sincosf((float)i * (6.283185307179586f / 1024.0f), &s, &cc);
        v2f w; w.x = cc; w.y = s;
        Tw[i] = w;
    }

    // ---- load shared input segment (zero-pad past TIME; padded region never
    //      feeds a valid window) ----
    const int segStart = g * (WPB * STRIDE);
    for (int i = tid; i < SEG; i += 128) {
        const int src = segStart + i;
        seg[i] = (src < TIME) ? x[(size_t)b * TIME + src] : 0.0f;
    }
    __syncthreads();

    const int wave = tid >> 5;
    const int lane = tid & 31;
    const int win  = g * WPB + wave;

    if (win < N_WIN) {                       // wave-uniform: EXEC stays all-ones
        const float* __restrict__ Xs = &seg[wave * STRIDE]; // X[n1][n2] = Xs[n1*32+n2]
        float* __restrict__ ar = Ar[wave];
        float* __restrict__ ai = Ai[wave];

        const int half = lane >> 4;          // lane half (0/1)
        const int lr   = lane & 15;

        // ================= Stage 1: A = (C - iS) * X =================
        // M=k1 (rows of C/S), K=n1, N=n2.  Store Ai negated => true imag part.
        #pragma unroll
        for (int mt = 0; mt < 2; ++mt)
            #pragma unroll
            for (int nt = 0; nt < 2; ++nt) {
                v8f accR = {}; v8f accI = {};
                #pragma unroll
                for (int ks = 0; ks < 8; ++ks) {
                    const int kb   = ks * 4 + half * 2;
                    const int arow = mt * 16 + lr;
                    const int bcol = nt * 16 + lr;
                    v2f aC = { Cm[arow * 32 + kb], Cm[arow * 32 + kb + 1] };
                    v2f aS = { Sm[arow * 32 + kb], Sm[arow * 32 + kb + 1] };
                    v2f bX = { Xs[kb * 32 + bcol], Xs[(kb + 1) * 32 + bcol] };
                    accR = WMMA_F32(aC, bX, accR);
                    accI = WMMA_F32(aS, bX, accI);
                }
                #pragma unroll
                for (int j = 0; j < 8; ++j) {
                    const int row = mt * 16 + j + half * 8;   // k1
                    const int col = nt * 16 + lr;             // n2
                    ar[row * 32 + col] =  accR[j];
                    ai[row * 32 + col] = -accI[j];
                }
            }

        // ================= Stage 2: twiddle B = A * e^{-2pi i k1 n2/1024} =====
        // table lookup: p = k1*n2 <= 31*31 < 1024
        for (int idx = lane; idx < 1024; idx += 32) {
            const int k1 = idx >> 5, n2 = idx & 31;
            const v2f w = Tw[k1 * n2];                 // {cos, sin}
            const float rr = ar[idx], ii = ai[idx];
            ar[idx] = rr * w.x + ii * w.y;
            ai[idx] = ii * w.x - rr * w.y;
        }

        // ====== Stage 3 (transposed): Xout^T = (C - iS) * B^T ================
        // M=k2 (A-operand from Cm/Sm, symmetric), K=n2, N=k1 (B-operand = B^T
        // read along contiguous rows of ar/ai).  D-tile N-index = k1 = fast
        // output index => coalesced stores.
        const size_t obase = ((size_t)b * N_WIN + win) * (size_t)KSIZE;
        v2f* __restrict__ out2 = (v2f*)out;

        #pragma unroll
        for (int mt = 0; mt < 2; ++mt)              // k2 tile
            #pragma unroll
            for (int nt = 0; nt < 2; ++nt) {        // k1 tile
                v8f aCR = {}, aSI = {}, aCI = {}, aSR = {};
                #pragma unroll
                for (int ks = 0; ks < 8; ++ks) {
                    const int kb   = ks * 4 + half * 2;
                    const int arow = mt * 16 + lr;  // k2 row of C/S
                    const int bcol = nt * 16 + lr;  // k1 column of B^T
                    v2f fC  = { Cm[arow * 32 + kb], Cm[arow * 32 + kb + 1] };
                    v2f fS  = { Sm[arow * 32 + kb], Sm[arow * 32 + kb + 1] };
                    v2f fBr = { ar[bcol * 32 + kb], ar[bcol * 32 + kb + 1] };
                    v2f fBi = { ai[bcol * 32 + kb], ai[bcol * 32 + kb + 1] };
                    aCR = WMMA_F32(fC, fBr, aCR);   // C * Br^T
                    aSI = WMMA_F32(fS, fBi, aSI);   // S * Bi^T
                    aCI = WMMA_F32(fC, fBi, aCI);   // C * Bi^T
                    aSR = WMMA_F32(fS, fBr, aSR);   // S * Br^T
                }
                #pragma unroll
                for (int j = 0; j < 8; ++j) {
                    const int k2 = mt * 16 + j + half * 8;
                    const int k1 = nt * 16 + lr;
                    v2f v;
                    v.x = aCR[j] + aSI[j];          // Re
                    v.y = aCI[j] - aSR[j];          // Im
                    __builtin_nontemporal_store(v, &out2[obase + (size_t)(k1 + 32 * k2)]);
                }
            }
    }
}

extern "C" void kernel_launch(void* const* d_in, const int* in_sizes, int n_in,
                              void* d_out, int out_size, void* d_ws, size_t ws_size,
                              hipStream_t stream) {
    (void)in_sizes; (void)n_in; (void)d_ws; (void)ws_size; (void)out_size;
    const float* x = (const float*)d_in[0];
    float* out = (float*)d_out;
    dim3 grid(BATCH * GROUPS);   // 32 * 256 = 8192 blocks
    dim3 block(128);             // 4 wave32 waves = 4 windows per block
    StridedFourier_kernel<<<grid, block, 0, stream>>>(x, out);
}